// Attention_21354577396085
// MI455X (gfx1250) — compile-verified
//
#include <hip/hip_runtime.h>
#include <math.h>

#define B_ 4
#define L_ 512
#define D_ 512
#define U_ 64
#define EPSV 1e-7f

typedef float v2f __attribute__((ext_vector_type(2)));
typedef float v8f __attribute__((ext_vector_type(8)));

// Native gfx1250 tanh (TRANS op, co-executes with VALU FMA stream).
static __device__ __forceinline__ float fast_tanh(float v) {
#if __has_builtin(__builtin_amdgcn_tanhf)
  return __builtin_amdgcn_tanhf(v);
#else
  float r;
  asm volatile("v_tanh_f32 %0, %1" : "=v"(r) : "v"(v));
  return r;
#endif
}

#define LOG2E_F 1.4426950408889634f

// ---------------------------------------------------------------------------
// Kernel 1: projections Q = X @ Wt, K = X @ Wx  (f32 WMMA 16x16x4)
// grid (B, L/16), block 128 (4 waves); wave w covers N-tile w of U=64.
// One A (x) load feeds two WMMAs per K-step.
// ---------------------------------------------------------------------------
__global__ __launch_bounds__(128) void proj_kernel(
    const float* __restrict__ x, const float* __restrict__ Wt,
    const float* __restrict__ Wx, float* __restrict__ q,
    float* __restrict__ kout_) {
  const int b    = blockIdx.x;
  const int i0   = blockIdx.y * 16;
  const int wave = threadIdx.x >> 5;
  const int lane = threadIdx.x & 31;
  const int n0   = wave * 16;

  const int m    = lane & 15;          // A row index
  const int koff = (lane >> 4) * 2;    // K sub-offset for this half-wave
  const int n    = lane & 15;          // B/D col index

  const float* arow = x + ((size_t)b * L_ + (i0 + m)) * D_;

  v8f cq = {};
  v8f ck = {};
  for (int kb = 0; kb < D_; kb += 4) {
    v2f a;
    a.x = arow[kb + koff];
    a.y = arow[kb + koff + 1];
    v2f bt, bx;
    bt.x = Wt[(size_t)(kb + koff) * U_ + n0 + n];
    bt.y = Wt[(size_t)(kb + koff + 1) * U_ + n0 + n];
    bx.x = Wx[(size_t)(kb + koff) * U_ + n0 + n];
    bx.y = Wx[(size_t)(kb + koff + 1) * U_ + n0 + n];
    cq = __builtin_amdgcn_wmma_f32_16x16x4_f32(false, a, false, bt, (short)0,
                                               cq, false, false);
    ck = __builtin_amdgcn_wmma_f32_16x16x4_f32(false, a, false, bx, (short)0,
                                               ck, false, false);
  }

  const int mbase = (lane >> 4) * 8;   // C/D: lanes 16-31 hold M=8..15
  float* qo = q + (size_t)b * L_ * U_;
  float* ko = kout_ + (size_t)b * L_ * U_;
#pragma unroll
  for (int r = 0; r < 8; ++r) {
    const size_t row = (size_t)(i0 + mbase + r) * U_ + n0 + n;
    qo[row] = cq[r];
    ko[row] = ck[r];
  }
}

// ---------------------------------------------------------------------------
// Kernel 2: fused scores -- never materialize h[B,L,L,U] (saves ~268 MB).
// One block per (b, i) row: 256 threads, each handles 2 columns j.
// e(i,j) = exp(sigmoid(sum_u Wa[u]*tanh(q_i[u]+bh[u]+k_j[u]) + ba)) * m_i*m_j
// then row-normalize: a = e / (sum_j e + eps).
// Inner loop: b128 k loads + v_tanh_f32 + v_fmac, ~67M tanh total.
// ---------------------------------------------------------------------------
__global__ __launch_bounds__(256) void score_kernel(
    const float* __restrict__ q, const float* __restrict__ kmat,
    const float* __restrict__ bh, const float* __restrict__ Wa,
    const float* __restrict__ ba, const unsigned char* __restrict__ mask,
    float* __restrict__ attn) {
  const int b = blockIdx.y;
  const int i = blockIdx.x;
  __shared__ float qb[U_];
  __shared__ float wa[U_];
  __shared__ float red[256];

  const int t = threadIdx.x;
  if (t < U_) {
    qb[t] = q[((size_t)b * L_ + i) * U_ + t] + bh[t];
    wa[t] = Wa[t];
  }
  __syncthreads();

  const float bav = ba[0];
  const float mi  = (float)mask[(size_t)b * L_ + i];

  float e[2];
#pragma unroll
  for (int p = 0; p < 2; ++p) {
    const int j = t + p * 256;
    // k rows live in our workspace at 256B-aligned offsets -> float4 safe.
    const float4* krow4 =
        (const float4*)(kmat + ((size_t)b * L_ + j) * U_);
    float s = 0.f;
#pragma unroll
    for (int u4 = 0; u4 < U_ / 4; ++u4) {
      const float4 kv = krow4[u4];
      s += wa[4 * u4 + 0] * fast_tanh(qb[4 * u4 + 0] + kv.x);
      s += wa[4 * u4 + 1] * fast_tanh(qb[4 * u4 + 1] + kv.y);
      s += wa[4 * u4 + 2] * fast_tanh(qb[4 * u4 + 2] + kv.z);
      s += wa[4 * u4 + 3] * fast_tanh(qb[4 * u4 + 3] + kv.w);
    }
    // sigmoid via native exp2/rcp, then exp via native exp2.
    const float z   = __builtin_amdgcn_exp2f(-(s + bav) * LOG2E_F);
    const float sig = __builtin_amdgcn_rcpf(1.0f + z);
    const float mj  = (float)mask[(size_t)b * L_ + j];
    e[p] = __builtin_amdgcn_exp2f(sig * LOG2E_F) * mi * mj;
  }

  red[t] = e[0] + e[1];
  __syncthreads();
  for (int off = 128; off > 0; off >>= 1) {
    if (t < off) red[t] += red[t + off];
    __syncthreads();
  }
  const float denom = red[0] + EPSV;

  float* arow = attn + ((size_t)b * L_ + i) * L_;
#pragma unroll
  for (int p = 0; p < 2; ++p) {
    arow[t + p * 256] = e[p] / denom;
  }
}

// ---------------------------------------------------------------------------
// Kernel 3: v = a @ x  (f32 WMMA 16x16x4, K = L = 512)
// grid (B, L/16, D/64), block 128 (4 waves), wave w covers 16 output cols.
// ---------------------------------------------------------------------------
__global__ __launch_bounds__(128) void av_kernel(
    const float* __restrict__ attn, const float* __restrict__ x,
    float* __restrict__ out) {
  const int b    = blockIdx.x;
  const int i0   = blockIdx.y * 16;
  const int wave = threadIdx.x >> 5;
  const int lane = threadIdx.x & 31;
  const int n0   = blockIdx.z * 64 + wave * 16;

  const int m    = lane & 15;
  const int koff = (lane >> 4) * 2;
  const int n    = lane & 15;

  const float* arow = attn + ((size_t)b * L_ + (i0 + m)) * L_;
  const float* xb   = x + (size_t)b * L_ * D_;

  v8f c = {};
  for (int kb = 0; kb < L_; kb += 4) {
    v2f a;
    a.x = arow[kb + koff];
    a.y = arow[kb + koff + 1];
    v2f bv;
    bv.x = xb[(size_t)(kb + koff) * D_ + n0 + n];
    bv.y = xb[(size_t)(kb + koff + 1) * D_ + n0 + n];
    c = __builtin_amdgcn_wmma_f32_16x16x4_f32(false, a, false, bv, (short)0, c,
                                              false, false);
  }

  const int mbase = (lane >> 4) * 8;
#pragma unroll
  for (int r = 0; r < 8; ++r) {
    out[((size_t)b * L_ + (i0 + mbase + r)) * D_ + n0 + n] = c[r];
  }
}

// ---------------------------------------------------------------------------
extern "C" void kernel_launch(void* const* d_in, const int* in_sizes, int n_in,
                              void* d_out, int out_size, void* d_ws,
                              size_t ws_size, hipStream_t stream) {
  (void)in_sizes; (void)n_in; (void)out_size; (void)ws_size;
  const float*         x    = (const float*)d_in[0];
  const unsigned char* mask = (const unsigned char*)d_in[1];  // bool -> u8
  const float*         Wt   = (const float*)d_in[2];
  const float*         Wx   = (const float*)d_in[3];
  const float*         bh   = (const float*)d_in[4];
  const float*         Wa   = (const float*)d_in[5];
  const float*         ba   = (const float*)d_in[6];
  float*               out  = (float*)d_out;

  float* q    = (float*)d_ws;                 // [B, L, U]
  float* k    = q + (size_t)B_ * L_ * U_;     // [B, L, U]
  float* attn = k + (size_t)B_ * L_ * U_;     // [B, L, L]

  proj_kernel<<<dim3(B_, L_ / 16), 128, 0, stream>>>(x, Wt, Wx, q, k);
  score_kernel<<<dim3(L_, B_), 256, 0, stream>>>(q, k, bh, Wa, ba, mask, attn);
  av_kernel<<<dim3(B_, L_ / 16, D_ / 64), 128, 0, stream>>>(attn, x, out);
}